// Mamba_58420145160299
// MI455X (gfx1250) — compile-verified
//
#include <hip/hip_runtime.h>
#include <hip/hip_bf16.h>
#include <math.h>

// ---------------------------------------------------------------------------
// Problem constants (match reference)
// ---------------------------------------------------------------------------
static constexpr int B_   = 2;
static constexpr int L    = 1024;
static constexpr int E    = 1024;
static constexpr int DIN  = 2 * E;          // 2048
static constexpr int N_ST = 16;             // SSM state size
static constexpr int R    = 64;
static constexpr int KCV  = 4;              // conv kernel width
static constexpr int M_ROWS = B_ * L;       // 2048 token rows
static constexpr int XZW  = 2 * DIN;        // 4096 (xi | z)
static constexpr int DBCW = R + 2 * N_ST;   // 96

typedef float v2f __attribute__((ext_vector_type(2)));
typedef float v8f __attribute__((ext_vector_type(8)));

static constexpr int BM = 64;   // block M tile (4 waves x 16 rows)
static constexpr int BN = 64;   // block N tile (4 sub-tiles of 16 per wave)
static constexpr int KC = 16;   // K chunk staged in LDS

// ---------------------------------------------------------------------------
// fp32 WMMA GEMM:  C[M x N] = A[M x K] * Bw[N x K]^T   (+ optional epilogue)
//
// Block = 128 threads (4 waves) -> 64x64 output tile.
//   wave w: rows [m0+16w, m0+16w+16), 4 accumulators across 64 columns.
// B chunk (64 x 16 fp32, 4KB) is staged in LDS via async DMA
// (GLOBAL_LOAD_ASYNC_TO_LDS_B128, tracked by ASYNCcnt), then consumed with
// ds_load_b64 fragments feeding V_WMMA_F32_16X16X4_F32.
//
// EPI == 0 : plain store
// EPI == 1 : C = softplus(acc + bias[n])      (delta path)
// ---------------------------------------------------------------------------
template <int EPI>
__global__ __launch_bounds__(128)
void mamba_wmma_gemm_f32(const float* __restrict__ A,
                         const float* __restrict__ Bw,
                         const float* __restrict__ bias,
                         float* __restrict__ C,
                         int Nn, int Kd, int lda, int ldb, int ldc) {
    __shared__ float tileB[BN * KC];    // 4 KB

    const int tid  = threadIdx.x;
    const int wave = tid >> 5;
    const int lane = tid & 31;
    const int half = lane >> 4;         // selects K pair {2h, 2h+1}
    const int l16  = lane & 15;

    const int m0 = blockIdx.x * BM;
    const int n0 = blockIdx.y * BN;

    // A: lane l16 = row within the wave's 16-row slice, contiguous float2 in K
    const float* arow = A + (size_t)(m0 + wave * 16 + l16) * lda + 2 * half;

    // Async B staging assignment: 4 lanes per row (16B each), 2 rows/thread.
    const int brow = tid >> 2;                 // 0..31
    const int bseg = (tid & 3) * 4;            // float offset within K chunk
    int gr0 = n0 + brow;      if (gr0 > Nn - 1) gr0 = Nn - 1;   // clamp (N=96 case)
    int gr1 = n0 + brow + 32; if (gr1 > Nn - 1) gr1 = Nn - 1;
    const float* bsrc0 = Bw + (size_t)gr0 * ldb + bseg;
    const float* bsrc1 = Bw + (size_t)gr1 * ldb + bseg;
    // LDS byte offsets (generic-addr low 32 bits == LDS address)
    const uint32_t ldst0 = (uint32_t)(uintptr_t)&tileB[(brow)      * KC + bseg];
    const uint32_t ldst1 = (uint32_t)(uintptr_t)&tileB[(brow + 32) * KC + bseg];

    v8f acc[4] = {};

    for (int k = 0; k < Kd; k += KC) {
        // ---- stage 64x16 B chunk into LDS with the async DMA path ----
        asm volatile("global_load_async_to_lds_b128 %0, %1, off"
                     :: "v"(ldst0), "v"(bsrc0 + k) : "memory");
        asm volatile("global_load_async_to_lds_b128 %0, %1, off"
                     :: "v"(ldst1), "v"(bsrc1 + k) : "memory");
        if (k + KC < Kd)
            __builtin_prefetch(arow + k + KC, 0, 3);   // near-cache A prefetch
        asm volatile("s_wait_asynccnt 0" ::: "memory");
        __syncthreads();

        // ---- 16 WMMAs per chunk: 4 K-steps x 4 N sub-tiles ----
#pragma unroll
        for (int kk = 0; kk < KC; kk += 4) {
            v2f a = *(const v2f*)(arow + k + kk);
#pragma unroll
            for (int j = 0; j < 4; ++j) {
                v2f b = *(const v2f*)&tileB[(j * 16 + l16) * KC + kk + 2 * half];
                // (neg_a, A, neg_b, B, c_mod, C, reuse_a, reuse_b)
                acc[j] = __builtin_amdgcn_wmma_f32_16x16x4_f32(
                    false, a, false, b, (short)0, acc[j], false, false);
            }
        }
        __syncthreads();
    }

    // ---- epilogue: D row = 16*wave + 8*half + v, col = n0 + 16j + l16 ----
    const int rbase = m0 + wave * 16 + half * 8;
#pragma unroll
    for (int j = 0; j < 4; ++j) {
        int col = n0 + j * 16 + l16;
        if (col < Nn) {
            float bn = (EPI == 1) ? bias[col] : 0.0f;
#pragma unroll
            for (int v = 0; v < 8; ++v) {
                float val = acc[j][v];
                if (EPI == 1) {
                    float t = val + bn;
                    val = (t > 20.0f) ? t : log1pf(__expf(t));  // stable softplus
                }
                C[(size_t)(rbase + v) * ldc + col] = val;
            }
        }
    }
}

// ---------------------------------------------------------------------------
// Depthwise causal conv (K=4) + bias + SiLU.  xi = xz[..., :DIN]
// xs[b,l,d] = silu( sum_k w[d,k] * xi[b, l-3+k, d] + b[d] )
// ---------------------------------------------------------------------------
__global__ __launch_bounds__(256)
void mamba_conv_silu(const float* __restrict__ xz,
                     const float* __restrict__ convw,
                     const float* __restrict__ convb,
                     float* __restrict__ xs) {
    int idx = blockIdx.x * blockDim.x + threadIdx.x;       // B_*L*DIN
    if (idx >= B_ * L * DIN) return;
    int d = idx & (DIN - 1);
    int l = (idx >> 11) & (L - 1);
    int b = idx >> 21;

    float acc = convb[d];
#pragma unroll
    for (int k = 0; k < KCV; ++k) {
        int ls = l - (KCV - 1) + k;
        if (ls >= 0)
            acc += convw[d * KCV + k] * xz[((size_t)(b * L + ls)) * XZW + d];
    }
    float s = acc / (1.0f + __expf(-acc));                 // silu
    xs[(size_t)(b * L + l) * DIN + d] = s;
}

// ---------------------------------------------------------------------------
// Sequential selective scan fused with y = hs.C + Dp*xs and u = y*silu(z).
// One lane per (b, d, n) state; 16-lane shfl_xor tree reduces over the 16
// SSM states inside each wave32 half. 65536 lanes total.
// ---------------------------------------------------------------------------
__global__ __launch_bounds__(256)
void mamba_scan(const float* __restrict__ delta,
                const float* __restrict__ dbc,
                const float* __restrict__ xs,
                const float* __restrict__ xz,     // z = xz[..., DIN:]
                const float* __restrict__ A_log,
                const float* __restrict__ Dp,
                float* __restrict__ u) {
    int gid = blockIdx.x * blockDim.x + threadIdx.x;       // B_*DIN*N_ST
    int n = gid & (N_ST - 1);
    int d = (gid >> 4) & (DIN - 1);
    int b = gid >> 15;

    const float Acoef = -__expf(A_log[d * N_ST + n]);      // A = -exp(A_log)
    const float dpv   = Dp[d];

    float h = 0.0f;
    for (int l = 0; l < L; ++l) {
        size_t row = (size_t)(b * L + l);
        float dlt = delta[row * DIN + d];                  // broadcast 16 lanes
        float xv  = xs[row * DIN + d];
        float Bv  = dbc[row * DBCW + R + n];
        float Cv  = dbc[row * DBCW + R + N_ST + n];

        h = __expf(dlt * Acoef) * h + dlt * Bv * xv;       // recurrence
        float t = h * Cv;
        t += __shfl_xor(t, 1, 32);   // xor masks 1..8 stay within 16-group
        t += __shfl_xor(t, 2, 32);
        t += __shfl_xor(t, 4, 32);
        t += __shfl_xor(t, 8, 32);
        if (n == 0) {
            float y  = t + dpv * xv;
            float zz = xz[row * XZW + DIN + d];
            float sz = zz / (1.0f + __expf(-zz));
            u[row * DIN + d] = y * sz;
        }
    }
}

// ---------------------------------------------------------------------------
// Host-side launcher
// ---------------------------------------------------------------------------
extern "C" void kernel_launch(void* const* d_in, const int* in_sizes, int n_in,
                              void* d_out, int out_size, void* d_ws, size_t ws_size,
                              hipStream_t stream) {
    const float* x      = (const float*)d_in[0];
    const float* W_in   = (const float*)d_in[1];
    const float* conv_w = (const float*)d_in[2];
    const float* conv_b = (const float*)d_in[3];
    const float* Wx     = (const float*)d_in[4];
    const float* Wdt    = (const float*)d_in[5];
    const float* bdt    = (const float*)d_in[6];
    const float* A_log  = (const float*)d_in[7];
    const float* Dp     = (const float*)d_in[8];
    const float* Wout   = (const float*)d_in[9];
    float* out = (float*)d_out;

    // workspace layout (floats)
    float* ws    = (float*)d_ws;
    float* xz    = ws;                                  // 2048*4096
    float* xs    = xz    + (size_t)M_ROWS * XZW;        // 2048*2048
    float* dbc   = xs    + (size_t)M_ROWS * DIN;        // 2048*96
    float* delta = dbc   + (size_t)M_ROWS * DBCW;       // 2048*2048
    float* u     = delta + (size_t)M_ROWS * DIN;        // 2048*2048

    dim3 blk(128);

    // 1) xz = x @ W_in^T        (2048 x 4096, K=1024)
    mamba_wmma_gemm_f32<0><<<dim3(M_ROWS / BM, XZW / BN), blk, 0, stream>>>(
        x, W_in, nullptr, xz, XZW, E, E, E, XZW);

    // 2) depthwise conv + SiLU -> xs
    mamba_conv_silu<<<(B_ * L * DIN) / 256, 256, 0, stream>>>(
        xz, conv_w, conv_b, xs);

    // 3) dbc = xs @ Wx^T        (2048 x 96, K=2048)   (N=96 -> guarded tiles)
    mamba_wmma_gemm_f32<0><<<dim3(M_ROWS / BM, (DBCW + BN - 1) / BN), blk, 0, stream>>>(
        xs, Wx, nullptr, dbc, DBCW, DIN, DIN, DIN, DBCW);

    // 4) delta = softplus(dbc[:, :R] @ Wdt^T + bdt)   (2048 x 2048, K=64)
    mamba_wmma_gemm_f32<1><<<dim3(M_ROWS / BM, DIN / BN), blk, 0, stream>>>(
        dbc, Wdt, bdt, delta, DIN, R, DBCW, R, DIN);

    // 5) selective scan + gating -> u
    mamba_scan<<<(B_ * DIN * N_ST) / 256, 256, 0, stream>>>(
        delta, dbc, xs, xz, A_log, Dp, u);

    // 6) out = u @ Wout^T       (2048 x 1024, K=2048)
    mamba_wmma_gemm_f32<0><<<dim3(M_ROWS / BM, E / BN), blk, 0, stream>>>(
        u, Wout, nullptr, out, E, DIN, DIN, DIN, E);
}